// diff_planner_ad_12601434046983
// MI455X (gfx1250) — compile-verified
//
#include <hip/hip_runtime.h>
#include <math.h>

// ---------------------------------------------------------------------------
// Batch ADMM trajectory solver for MI455X (gfx1250), full FP32 via
// V_WMMA_F32_16X16X4_F32 (K=4 chained x4 => K=16 contraction).
// One wave owns a 16-batch tile. Linear residual projections fold into
// constant Gram matrices; the 3 nonlinear x/y projections chain into a
// single WMMA accumulator per axis to keep VGPR count under 256.
// ---------------------------------------------------------------------------

#define NBATCH  4096
#define NUMT    128
#define NOBSN   10
#define MAXIT   30
#define NTILES  8      // 128 / 16

typedef __attribute__((ext_vector_type(2))) float v2f;
typedef __attribute__((ext_vector_type(8))) float v8f;

__device__ __forceinline__ v2f mk2(float a, float b) { v2f r; r.x = a; r.y = b; return r; }

// D(16x16,f32) = A(16x4,f32) * B(4x16,f32) + C   -- CDNA5 WMMA, wave32
__device__ __forceinline__ v8f wmma4(v2f a, v2f b, v8f c) {
  return __builtin_amdgcn_wmma_f32_16x16x4_f32(false, a, false, b, (short)0, c, false, false);
}

// Workspace layout (floats)
#define WS_FX   0      // cmi_x[:16,:16]   stored F[n][k]   (256)
#define WS_FY   256    // cmi_y[:16,:16]                    (256)
#define WS_GX   512    // cmi_x[:16,16:24] stored G[n][k]   (128)
#define WS_GY   640    // cmi_y[:16,16:25]                  (144)
#define WS_EX0  784    // cmi_x0[:16,16:24]                 (128)
#define WS_EY0  912    // cmi_y0[:16,16:25]                 (144)
#define WS_GSUM 1056   // Pdd^T Pdd + Pd^T Pd + 10 P^T P    (256)
#define WS_GP   1312   // P^T P                             (256)
#define WS_S1   1568   // column sums of P                  (16)
#define WS_ST   1584   // t-weighted column sums of P       (16)

// ---------------------------------------------------------------------------
// Setup: Gram matrices, 4 KKT inverses (Gauss-Jordan), column sums.
// ---------------------------------------------------------------------------
__device__ __forceinline__ float aeq_elem(bool ysys, int r, int col, const float sRow[5][16]) {
  if (!ysys) {
    if (r == 0) return sRow[0][col];           // P[0]
    if (r == 1) return sRow[1][col];           // Pdot[0]
    if (r == 2) return sRow[2][col];           // Pddot[0]
    if (r == 3) return sRow[4][col];           // Pdot[-1]
    return (col == r) ? 1.0f : 0.0f;           // rows 4..7 -> e4..e7
  } else {
    if (r == 0) return sRow[0][col];
    if (r == 1) return sRow[1][col];
    if (r == 2) return sRow[2][col];
    if (r == 3) return sRow[3][col];           // P[-1]
    if (r == 4) return sRow[4][col];           // Pdot[-1]
    return (col == r - 1) ? 1.0f : 0.0f;       // rows 5..8 -> e4..e7
  }
}

__global__ __launch_bounds__(256) void setup_kernel(const float* __restrict__ P,
                                                    const float* __restrict__ Pd,
                                                    const float* __restrict__ Pdd,
                                                    float* __restrict__ ws) {
  __shared__ float sG[3][16][16];   // P^T P, Pd^T Pd, Pdd^T Pdd
  __shared__ float sRow[5][16];
  __shared__ float sAug[25 * 52];
  __shared__ float sCol[25];
  const int tid = threadIdx.x;

  { // Gram matrices
    int i = tid >> 4, j = tid & 15;
    float a = 0.f, b = 0.f, c = 0.f;
    for (int t = 0; t < NUMT; t++) {
      a += P[t * 16 + i]  * P[t * 16 + j];
      b += Pd[t * 16 + i] * Pd[t * 16 + j];
      c += Pdd[t * 16 + i]* Pdd[t * 16 + j];
    }
    sG[0][i][j] = a; sG[1][i][j] = b; sG[2][i][j] = c;
  }
  if (tid < 16) {
    sRow[0][tid] = P[tid];  sRow[1][tid] = Pd[tid];  sRow[2][tid] = Pdd[tid];
    sRow[3][tid] = P[(NUMT - 1) * 16 + tid];
    sRow[4][tid] = Pd[(NUMT - 1) * 16 + tid];
    float a = 0.f, b = 0.f;
    for (int t = 0; t < NUMT; t++) {
      float tv = (float)t * (10.0f / 127.0f);
      a += P[t * 16 + tid];
      b += tv * P[t * 16 + tid];
    }
    ws[WS_S1 + tid] = a; ws[WS_ST + tid] = b;
  }
  __syncthreads();

  { // constant projection matrices
    int i = tid >> 4, j = tid & 15;
    ws[WS_GSUM + tid] = sG[2][i][j] + sG[1][i][j] + 10.0f * sG[0][i][j];
    ws[WS_GP + tid]   = sG[0][i][j];
  }

  for (int sys = 0; sys < 4; sys++) {
    const bool ysys = (sys == 1 || sys == 3);
    const bool full = (sys >= 2);
    const int n = ysys ? 25 : 24;
    for (int idx = tid; idx < n * 2 * n; idx += blockDim.x) {
      int i = idx / (2 * n), j = idx - i * (2 * n);
      float v;
      if (j >= n) {
        v = (j - n == i) ? 1.0f : 0.0f;
      } else if (i < 16 && j < 16) {
        float m = 100.0f * sG[2][i][j] + ((i == j) ? 10.0f : 0.0f);
        if (full) {
          m += sG[2][i][j] + sG[1][i][j] + 10.0f * sG[0][i][j];
          if (ysys) m += 2.0f * sG[0][i][j];
        }
        v = m;
      } else if (i >= 16 && j >= 16) {
        v = 0.0f;
      } else {
        int r   = (i >= 16) ? (i - 16) : (j - 16);
        int col = (i >= 16) ? j : i;
        v = aeq_elem(ysys, r, col, sRow);
      }
      sAug[i * 52 + j] = v;
    }
    for (int k = 0; k < n; k++) {           // Gauss-Jordan
      __syncthreads();
      float ipiv = 1.0f / sAug[k * 52 + k];
      if (tid < 2 * n) sAug[k * 52 + tid] *= ipiv;
      if (tid < n) sCol[tid] = sAug[tid * 52 + k];
      __syncthreads();
      for (int idx = tid; idx < n * 2 * n; idx += blockDim.x) {
        int i = idx / (2 * n), j = idx - i * (2 * n);
        if (i != k) sAug[i * 52 + j] -= sCol[i] * sAug[k * 52 + j];
      }
    }
    __syncthreads();
    if (sys == 0) {
      for (int idx = tid; idx < 16 * 8; idx += blockDim.x) {
        int r = idx >> 3, c = idx & 7;
        ws[WS_EX0 + idx] = sAug[r * 52 + n + 16 + c];
      }
    } else if (sys == 1) {
      for (int idx = tid; idx < 16 * 9; idx += blockDim.x) {
        int r = idx / 9, c = idx - r * 9;
        ws[WS_EY0 + idx] = sAug[r * 52 + n + 16 + c];
      }
    } else if (sys == 2) {
      for (int idx = tid; idx < 256; idx += blockDim.x) {
        int r = idx >> 4, c = idx & 15;
        ws[WS_FX + idx] = sAug[r * 52 + n + c];
      }
      for (int idx = tid; idx < 16 * 8; idx += blockDim.x) {
        int r = idx >> 3, c = idx & 7;
        ws[WS_GX + idx] = sAug[r * 52 + n + 16 + c];
      }
    } else {
      for (int idx = tid; idx < 256; idx += blockDim.x) {
        int r = idx >> 4, c = idx & 15;
        ws[WS_FY + idx] = sAug[r * 52 + n + c];
      }
      for (int idx = tid; idx < 16 * 9; idx += blockDim.x) {
        int r = idx / 9, c = idx - r * 9;
        ws[WS_GY + idx] = sAug[r * 52 + n + 16 + c];
      }
    }
    __syncthreads();
  }
}

// ---------------------------------------------------------------------------
// Main solver: 1 wave per block, 16 batches per wave, 30 iterations.
// ---------------------------------------------------------------------------
__global__ __launch_bounds__(32) void solve_kernel(
    const float* __restrict__ P,  const float* __restrict__ Pd,
    const float* __restrict__ Pdd,
    const float* __restrict__ init_state, const float* __restrict__ fin_state,
    const float* __restrict__ cobs, const float* __restrict__ vobs,
    const float* __restrict__ yub, const float* __restrict__ ylb,
    const float* __restrict__ lamx_g, const float* __restrict__ lamy_g,
    const float* __restrict__ cxp, const float* __restrict__ cyp,
    const float* __restrict__ ws, float* __restrict__ out) {
  __shared__ float sP[128][16], sPd[128][16], sPdd[128][16];
  __shared__ float sFx[16][16], sFy[16][16];
  __shared__ float sGs[16][16], sGq[16][16];     // Gsum, Gp
  __shared__ float sCxT[16][16], sCyT[16][16];   // coeffs, [nvar k][batch m]
  __shared__ float sS[7][16][16];                // nonlinear staging, [k][m]
  __shared__ float sOxT[10][16], sOyT[10][16], sOvxT[10][16], sOvyT[10][16];
  __shared__ float sUB[16], sLB[16];
  __shared__ float sBx[16][8], sBy[16][9];
  __shared__ float sHx[16][16], sHy[16][16];     // beq @ G^T          [n][m]
  __shared__ float sPXo[16][16], sPYo[16][16];   // proj(sum obs traj) [n][m]
  __shared__ float sO0[4][16];

  const int lane = threadIdx.x;
  const int ln = lane & 15;
  const int hi = lane >> 4;
  const int ko = hi * 2;            // fragment K-offset for hi half-wave
  const int base = blockIdx.x * 16;

  for (int i = lane; i < 128 * 16; i += 32) {
    int r = i >> 4, c = i & 15;
    sP[r][c] = P[i]; sPd[r][c] = Pd[i]; sPdd[r][c] = Pdd[i];
  }
  for (int i = lane; i < 256; i += 32) {
    int r = i >> 4, c = i & 15;
    sFx[r][c] = ws[WS_FX + i];   sFy[r][c] = ws[WS_FY + i];
    sGs[r][c] = ws[WS_GSUM + i]; sGq[r][c] = ws[WS_GP + i];
  }
  for (int b = lane; b < 16; b += 32) {
    int gb = base + b;
    float xi = init_state[gb * 4 + 0], yi = init_state[gb * 4 + 1];
    float vi = init_state[gb * 4 + 2], ps = init_state[gb * 4 + 3];
    sBx[b][0] = xi; sBx[b][1] = vi * cosf(ps); sBx[b][2] = 0.0f;
    sBx[b][3] = fin_state[gb * 3 + 0];
    for (int k = 0; k < 4; k++) sBx[b][4 + k] = cxp[gb * 4 + k];
    sBy[b][0] = yi; sBy[b][1] = vi * sinf(ps); sBy[b][2] = 0.0f;
    sBy[b][3] = fin_state[gb * 3 + 1]; sBy[b][4] = 0.0f;
    for (int k = 0; k < 4; k++) sBy[b][5 + k] = cyp[gb * 4 + k];
    sUB[b] = yub[gb]; sLB[b] = ylb[gb];
    float X0 = 0.f, Y0 = 0.f, VX0 = 0.f, VY0 = 0.f;
    for (int o = 0; o < NOBSN; o++) {
      float xo = cobs[gb * 20 + o], yo = cobs[gb * 20 + 10 + o];
      float vx = vobs[gb * 20 + o], vy = vobs[gb * 20 + 10 + o];
      sOxT[o][b] = xo; sOyT[o][b] = yo; sOvxT[o][b] = vx; sOvyT[o][b] = vy;
      X0 += xo; Y0 += yo; VX0 += vx; VY0 += vy;
    }
    sO0[0][b] = X0; sO0[1][b] = Y0; sO0[2][b] = VX0; sO0[3][b] = VY0;
  }
  __syncthreads();

  const float s1v = ws[WS_S1 + ln], stv = ws[WS_ST + ln];

  // persistent per-lane fragments (only the duals stay in registers)
  float lxr[8], lyr[8];
#pragma unroll
  for (int v = 0; v < 8; v++) {
    int m = v + 8 * hi, gb = base + m;
    lxr[v] = lamx_g[gb * 16 + ln];
    lyr[v] = lamy_g[gb * 16 + ln];
    float hx = 0.f, hy = 0.f, c0x = 0.f, c0y = 0.f;
    for (int k = 0; k < 8; k++) {
      float bv = sBx[m][k];
      hx  += bv * ws[WS_GX  + ln * 8 + k];
      c0x += bv * ws[WS_EX0 + ln * 8 + k];
    }
    for (int k = 0; k < 9; k++) {
      float bv = sBy[m][k];
      hy  += bv * ws[WS_GY  + ln * 9 + k];
      c0y += bv * ws[WS_EY0 + ln * 9 + k];
    }
    sHx[ln][m]  = hx;  sHy[ln][m]  = hy;
    sPXo[ln][m] = sO0[0][m] * s1v + sO0[2][m] * stv;
    sPYo[ln][m] = sO0[1][m] * s1v + sO0[3][m] * stv;
    sCxT[ln][m] = c0x; sCyT[ln][m] = c0y;          // initial primal
  }

#pragma unroll 1
  for (int it = 0; it < MAXIT; it++) {
    __syncthreads();
    v2f acx[4], acy[4];
#pragma unroll
    for (int c = 0; c < 4; c++) {
      int kk = 4 * c + ko;
      acx[c] = mk2(sCxT[kk][ln], sCxT[kk + 1][ln]);
      acy[c] = mk2(sCyT[kk][ln], sCyT[kk + 1][ln]);
    }
    // qX/qY accumulate ALL nonlinear back-projections for their axis; they
    // start at proj_P(sum_o obs_traj) so that qX == Sx after the tile loop.
    v8f qX, qY, qLr;
#pragma unroll
    for (int v = 0; v < 8; v++) {
      int m = v + 8 * hi;
      qX[v] = sPXo[ln][m]; qY[v] = sPYo[ln][m]; qLr[v] = 0.0f;
    }

#pragma unroll 1
    for (int T = 0; T < NTILES; T++) {
      const int t0 = T * 16;
      __syncthreads();            // previous tile's sS reads complete
      v2f bFp[4], bFd[4], bFdd[4];
#pragma unroll
      for (int c = 0; c < 4; c++) {
        int kk = 4 * c + ko;
        bFp[c]  = mk2(sP  [t0 + ln][kk], sP  [t0 + ln][kk + 1]);
        bFd[c]  = mk2(sPd [t0 + ln][kk], sPd [t0 + ln][kk + 1]);
        bFdd[c] = mk2(sPdd[t0 + ln][kk], sPdd[t0 + ln][kk + 1]);
      }
      // forward: x, xdot, xddot, y, ydot, yddot  (24 WMMA)
      v8f X = {}, Xd = {}, Xdd = {}, Y = {}, Yd = {}, Ydd = {};
#pragma unroll
      for (int v = 0; v < 8; v++) { X[v]=0.f; Xd[v]=0.f; Xdd[v]=0.f; Y[v]=0.f; Yd[v]=0.f; Ydd[v]=0.f; }
#pragma unroll
      for (int c = 0; c < 4; c++) {
        X   = wmma4(acx[c], bFp[c],  X);
        Xd  = wmma4(acx[c], bFd[c],  Xd);
        Xdd = wmma4(acx[c], bFdd[c], Xdd);
        Y   = wmma4(acy[c], bFp[c],  Y);
        Yd  = wmma4(acy[c], bFd[c],  Yd);
        Ydd = wmma4(acy[c], bFdd[c], Ydd);
      }
      const float tv = (float)(t0 + ln) * (10.0f / 127.0f);
      // obstacles: [o][m] layout -> contiguous 8-float reads per (o, array)
      float oqx[8], oqy[8];
#pragma unroll
      for (int v = 0; v < 8; v++) { oqx[v] = 0.f; oqy[v] = 0.f; }
      for (int o = 0; o < NOBSN; o++) {
#pragma unroll
        for (int v = 0; v < 8; v++) {
          int m = v + 8 * hi;
          float xo = sOxT[o][m] + sOvxT[o][m] * tv;
          float yo = sOyT[o][m] + sOvyT[o][m] * tv;
          float wc = X[v] - xo, wsn = Y[v] - yo;
          float rr2 = fmaxf((3.2f * wc) * (3.2f * wc) + (6.0f * wsn) * (6.0f * wsn), 1e-24f);
          float u = fmaxf(1.0f, 19.2f * rsqrtf(rr2));   // A*B/ellipse dist
          oqx[v] += u * wc; oqy[v] += u * wsn;
        }
      }
#pragma unroll
      for (int v = 0; v < 8; v++) {
        const int m = v + 8 * hi;
        float xd = Xd[v], xdd = Xdd[v], y = Y[v], yd = Yd[v], ydd = Ydd[v];
        float rv = sqrtf(fmaxf(xd * xd + yd * yd, 1e-24f));
        float wv = fminf(fmaxf(rv, 0.1f), 30.0f) / rv;        // d_v/|v|
        float ra = sqrtf(fmaxf(xdd * xdd + ydd * ydd, 1e-24f));
        float wa = fminf(ra, 18.0f) / ra;                     // d_a/|a|
        float ub = sUB[m], lb = sLB[m];
        sS[0][ln][m] = xdd * wa;                                // d_a*ca
        sS[1][ln][m] = xd * wv;                                 // d_v*cv
        sS[2][ln][m] = oqx[v];                                  // sum u*wc
        sS[3][ln][m] = ydd * wa;                                // d_a*sa
        sS[4][ln][m] = yd * wv;                                 // d_v*sv
        sS[5][ln][m] = oqy[v];                                  // sum u*ws
        sS[6][ln][m] = fmaxf(0.f, y - ub) - fmaxf(0.f, lb - y); // lane res
      }
      __syncthreads();
      // all nonlinear back-projections chained into qX / qY / qLr (28 WMMA)
      v2f bPp[4], bPd_[4], bPdd_[4];
#pragma unroll
      for (int c = 0; c < 4; c++) {
        int kk = 4 * c + ko;
        bPp[c]   = mk2(sP  [t0 + kk][ln], sP  [t0 + kk + 1][ln]);
        bPd_[c]  = mk2(sPd [t0 + kk][ln], sPd [t0 + kk + 1][ln]);
        bPdd_[c] = mk2(sPdd[t0 + kk][ln], sPdd[t0 + kk + 1][ln]);
      }
#pragma unroll
      for (int c = 0; c < 4; c++) {
        int kk = 4 * c + ko;
        qX  = wmma4(mk2(sS[0][kk][ln], sS[0][kk + 1][ln]), bPdd_[c], qX);
        qX  = wmma4(mk2(sS[1][kk][ln], sS[1][kk + 1][ln]), bPd_[c],  qX);
        qX  = wmma4(mk2(sS[2][kk][ln], sS[2][kk + 1][ln]), bPp[c],   qX);
        qY  = wmma4(mk2(sS[3][kk][ln], sS[3][kk + 1][ln]), bPdd_[c], qY);
        qY  = wmma4(mk2(sS[4][kk][ln], sS[4][kk + 1][ln]), bPd_[c],  qY);
        qY  = wmma4(mk2(sS[5][kk][ln], sS[5][kk + 1][ln]), bPp[c],   qY);
        qLr = wmma4(mk2(sS[6][kk][ln], sS[6][kk + 1][ln]), bPp[c],   qLr);
      }
    }

    // linear parts: c @ Gsum, c_y @ Gp (12 WMMA) -- after tile loop to keep
    // these accumulators (and the G fragments) out of the tile-loop live set
    v8f cxG, cyG, cyQ;
#pragma unroll
    for (int v = 0; v < 8; v++) { cxG[v] = 0.f; cyG[v] = 0.f; cyQ[v] = 0.f; }
#pragma unroll
    for (int c = 0; c < 4; c++) {
      int kk = 4 * c + ko;
      v2f bGs = mk2(sGs[kk][ln], sGs[kk + 1][ln]);
      v2f bGq = mk2(sGq[kk][ln], sGq[kk + 1][ln]);
      cxG = wmma4(acx[c], bGs, cxG);
      cyG = wmma4(acy[c], bGs, cyG);
      cyQ = wmma4(acy[c], bGq, cyQ);
    }

    // dual update + (-lincost):
    //   lx' = lx - cx@Gsum + Sx ;         -lincost_x = lx' + Sx        (Sx=qX)
    //   ly' = ly - cy@Gsum + Sy - pLr ;   -lincost_y = ly' + Sy + 2 cy@Gp - pLr
    float lcx[8], lcy[8];
#pragma unroll
    for (int v = 0; v < 8; v++) {
      float lxn = lxr[v] - cxG[v] + qX[v];
      lxr[v] = lxn; lcx[v] = lxn + qX[v];
      float lyn = lyr[v] - cyG[v] + qY[v] - qLr[v];
      lyr[v] = lyn; lcy[v] = lyn + qY[v] + 2.0f * cyQ[v] - qLr[v];
    }
    // KKT solve: c = (-lincost) @ F^T + h   (8 WMMA)
    __syncthreads();
#pragma unroll
    for (int v = 0; v < 8; v++) {
      int m = v + 8 * hi;
      sS[0][ln][m] = lcx[v]; sS[1][ln][m] = lcy[v];
    }
    __syncthreads();
    v8f ncx, ncy;
#pragma unroll
    for (int v = 0; v < 8; v++) {
      int m = v + 8 * hi;
      ncx[v] = sHx[ln][m]; ncy[v] = sHy[ln][m];
    }
#pragma unroll
    for (int c = 0; c < 4; c++) {
      int kk = 4 * c + ko;
      v2f ax = mk2(sS[0][kk][ln], sS[0][kk + 1][ln]);
      v2f ay = mk2(sS[1][kk][ln], sS[1][kk + 1][ln]);
      v2f bx = mk2(sFx[ln][kk], sFx[ln][kk + 1]);
      v2f by = mk2(sFy[ln][kk], sFy[ln][kk + 1]);
      ncx = wmma4(ax, bx, ncx);
      ncy = wmma4(ay, by, ncy);
    }
    __syncthreads();
#pragma unroll
    for (int v = 0; v < 8; v++) {
      int m = v + 8 * hi;
      sCxT[ln][m] = ncx[v]; sCyT[ln][m] = ncy[v];
    }
  }

  __syncthreads();
  // primal_sol = [cx | cy] per batch ; cx[m][k] lives at sCxT[k][m]
#pragma unroll
  for (int v = 0; v < 8; v++) {
    int m = v + 8 * hi, gb = base + m;
    out[gb * 32 + ln]      = sCxT[ln][m];
    out[gb * 32 + 16 + ln] = sCyT[ln][m];
  }
}

extern "C" void kernel_launch(void* const* d_in, const int* in_sizes, int n_in,
                              void* d_out, int out_size, void* d_ws, size_t ws_size,
                              hipStream_t stream) {
  (void)in_sizes; (void)n_in; (void)out_size; (void)ws_size;
  const float* P    = (const float*)d_in[0];
  const float* Pd   = (const float*)d_in[1];
  const float* Pdd  = (const float*)d_in[2];
  const float* ini  = (const float*)d_in[3];
  const float* fin  = (const float*)d_in[4];
  const float* cobs = (const float*)d_in[5];
  const float* vobs = (const float*)d_in[6];
  const float* yubp = (const float*)d_in[7];
  const float* ylbp = (const float*)d_in[8];
  const float* lamx = (const float*)d_in[9];
  const float* lamy = (const float*)d_in[10];
  const float* cxp  = (const float*)d_in[11];
  const float* cyp  = (const float*)d_in[12];
  float* ws  = (float*)d_ws;
  float* out = (float*)d_out;

  hipLaunchKernelGGL(setup_kernel, dim3(1), dim3(256), 0, stream, P, Pd, Pdd, ws);
  hipLaunchKernelGGL(solve_kernel, dim3(NBATCH / 16), dim3(32), 0, stream,
                     P, Pd, Pdd, ini, fin, cobs, vobs, yubp, ylbp, lamx, lamy,
                     cxp, cyp, ws, out);
}